// PagedKVCache_84920093377303
// MI455X (gfx1250) — compile-verified
//
#include <hip/hip_runtime.h>
#include <hip/hip_bf16.h>
#include <stdint.h>

// Native 128-bit vector (clang ext_vector, accepted by nontemporal builtins).
typedef __attribute__((ext_vector_type(4))) unsigned int v4u;

// Problem constants (match reference setup_inputs()).
#define BB 8
#define SS 2048
#define HH 8
#define DD 128
#define PAGE_SIZE 16
#define N_PAGES 2048
#define MAX_BLOCKS (SS / PAGE_SIZE)   // 128

// Elements/bytes per cache tensor: N_PAGES * H * PAGE_SIZE * D
#define CACHE_ELEMS ((size_t)N_PAGES * HH * PAGE_SIZE * DD)   // 33,554,432 bf16
#define CACHE_BYTES (CACHE_ELEMS * 2)                          // 67,108,864 B

// ---------------------------------------------------------------------------
// Kernel 1: bulk streaming copy src -> dst using the gfx1250 async
// global<->LDS DMA path (ASYNCcnt). Each block moves 32 KiB:
//   256 threads x 8 slots x 16 B. Loads are issued 8-deep (they pipeline and
//   complete in order among themselves), one s_wait_asynccnt 0, then 8 async
//   stores, one final wait. No VGPR data movement at all.
// ---------------------------------------------------------------------------
__global__ void __launch_bounds__(256)
bulk_copy_async_kernel(const char* __restrict__ src, char* __restrict__ dst) {
    __shared__ char smem[32 * 1024];

    const unsigned tid  = threadIdx.x;             // 0..255
    const size_t   base = (size_t)blockIdx.x * (32u * 1024u);

    const char* s = src + base + (size_t)tid * 16u;
    char*       d = dst + base + (size_t)tid * 16u;

    // LDS byte offset of this thread's slot 0 (AS(3) pointer value == LDS offset).
    const unsigned lds0 = (unsigned)(size_t)(&smem[0]) + tid * 16u;

#pragma unroll
    for (int j = 0; j < 8; ++j) {
        const unsigned l  = lds0 + (unsigned)j * 4096u;   // slot j region
        const char*    sa = s + (size_t)j * 4096u;
        asm volatile("global_load_async_to_lds_b128 %0, %1, off"
                     :: "v"(l), "v"(sa)
                     : "memory");
    }
    // All 8 async loads complete -> LDS slots valid.
    asm volatile("s_wait_asynccnt 0x0" ::: "memory");

#pragma unroll
    for (int j = 0; j < 8; ++j) {
        const unsigned l  = lds0 + (unsigned)j * 4096u;
        char*          da = d + (size_t)j * 4096u;
        asm volatile("global_store_async_from_lds_b128 %0, %1, off"
                     :: "v"(da), "v"(l)
                     : "memory");
    }
    // Drain stores before wave exit (S_ENDPGM also implies wait-idle).
    asm volatile("s_wait_asynccnt 0x0" ::: "memory");
}

// ---------------------------------------------------------------------------
// Kernel 2: paged scatter of k_val / v_val into the (already copied) output
// caches. One block per token (b, s); 256 threads cover
//   {K,V} x 8 heads x 16 chunks of 16 B  (= 2 x 8 x 128 bf16 = 4 KiB/token).
// All accesses are 128-bit and coalesced; NT hints keep the streamed data
// from displacing L2 (nothing re-reads it on-device).
// ---------------------------------------------------------------------------
__global__ void __launch_bounds__(256)
scatter_kv_kernel(const v4u* __restrict__ k_val,
                  const v4u* __restrict__ v_val,
                  const int* __restrict__ page_table,
                  const int* __restrict__ input_pos,
                  const int* __restrict__ batch_idx,
                  v4u* __restrict__ out_k,
                  v4u* __restrict__ out_v) {
    const int token = blockIdx.x;          // 0 .. B*S-1
    const int b     = token >> 11;         // / SS (SS == 2048)
    const int s     = token & (SS - 1);

    const int pos  = input_pos[token];     // input_pos[b, s]
    const int blk  = pos >> 4;             // / PAGE_SIZE
    const int slot = pos & (PAGE_SIZE - 1);
    const int bi   = batch_idx[b];
    const int page = page_table[bi * MAX_BLOCKS + blk];

    const int tid   = threadIdx.x;         // 0..255
    const int cache = tid >> 7;            // 0 = K, 1 = V
    const int r     = tid & 127;
    const int h     = r >> 4;              // head 0..7
    const int c     = r & 15;              // 16B chunk within 256B row (D/8 = 16)

    const v4u* __restrict__ src = cache ? v_val : k_val;
    v4u* __restrict__       dst = cache ? out_v : out_k;

    // src: [B, H, S, D] row (b, h, s, :) -> 16 x 16B chunks
    const size_t si = ((size_t)((b * HH + h) * SS + s)) * (DD / 8) + c;
    // dst: [N_PAGES, H, PAGE_SIZE, D] row (page, h, slot, :) -> 16 x 16B chunks
    const size_t di = ((size_t)((page * HH + h) * PAGE_SIZE + slot)) * (DD / 8) + c;

    const v4u v = __builtin_nontemporal_load(&src[si]);
    __builtin_nontemporal_store(v, &dst[di]);
}

// ---------------------------------------------------------------------------
// Launch: copy both caches into d_out (async-DMA memcpy), then scatter the
// new tokens on top. Same stream -> kernels serialize, so scatter wins.
// ---------------------------------------------------------------------------
extern "C" void kernel_launch(void* const* d_in, const int* in_sizes, int n_in,
                              void* d_out, int out_size, void* d_ws, size_t ws_size,
                              hipStream_t stream) {
    (void)in_sizes; (void)n_in; (void)out_size; (void)d_ws; (void)ws_size;

    const char* k_cache = (const char*)d_in[0];
    const char* v_cache = (const char*)d_in[1];
    const v4u* k_val    = (const v4u*)d_in[2];
    const v4u* v_val    = (const v4u*)d_in[3];
    const int* page_table = (const int*)d_in[4];
    const int* input_pos  = (const int*)d_in[5];
    const int* batch_idx  = (const int*)d_in[6];

    __hip_bfloat16* out   = (__hip_bfloat16*)d_out;
    char*           out_k = (char*)out;
    char*           out_v = (char*)(out + CACHE_ELEMS);

    // 64 MiB per cache / 32 KiB per block = 2048 blocks per cache.
    const int copy_blocks = (int)(CACHE_BYTES / (32 * 1024));
    bulk_copy_async_kernel<<<copy_blocks, 256, 0, stream>>>(k_cache, out_k);
    bulk_copy_async_kernel<<<copy_blocks, 256, 0, stream>>>(v_cache, out_v);

    // One block per token.
    scatter_kv_kernel<<<BB * SS, 256, 0, stream>>>(
        k_val, v_val, page_table, input_pos, batch_idx,
        (v4u*)out_k, (v4u*)out_v);
}